// MedianBlur_84602265796897
// MI455X (gfx1250) — compile-verified
//
#include <hip/hip_runtime.h>
#include <stdint.h>

// Problem constants (fixed by the harness: [32, 3, 512, 512] f32).
#define W        512
#define H        512
#define TILE_H   8                   // output rows per workgroup
#define TILESY   (H / TILE_H)        // 64 vertical tiles per plane
#define ROWS     (TILE_H + 2)        // staged rows incl. top/bottom halo
#define LDSW     520                 // padded LDS row stride (floats); 2080 B, 16B-aligned rows
#define SEGS     (W / 4)             // 128 float4 segments per row
#define NTHREADS 256

typedef float v4f __attribute__((ext_vector_type(4)));

__device__ __forceinline__ void sort3(float a, float b, float c,
                                      float& mn, float& md, float& mx) {
  float lo = fminf(a, b);
  float hi = fmaxf(a, b);
  mx = fmaxf(hi, c);
  md = fmaxf(fminf(hi, c), lo);
  mn = fminf(lo, c);
}
__device__ __forceinline__ float med3(float a, float b, float c) {
  return fmaxf(fminf(a, b), fminf(fmaxf(a, b), c));
}

__global__ __launch_bounds__(NTHREADS)
void MedianBlur3x3_kernel(const float* __restrict__ in,
                          float* __restrict__ out) {
  __shared__ float tile[ROWS * LDSW];   // 20.8 KB of 320 KB/WGP

  const int tid = threadIdx.x;
  const int bc  = blockIdx.x >> 6;      // (batch*channel) plane, 0..95
  const int ty  = blockIdx.x & (TILESY - 1);
  const int y0  = ty * TILE_H;

  const float* plane = in + (size_t)bc * (H * W);

  // ---- Stage ROWS x 512 floats into LDS via CDNA5 async global->LDS B128 DMA.
  // Interior data lands at column offset 4 (16B aligned); halo cols at 3 and 516.
  const uint32_t tile_base = (uint32_t)(uintptr_t)(&tile[0]); // low 32b of flat = LDS offset
  #pragma unroll
  for (int i = 0; i < (ROWS * SEGS) / NTHREADS; ++i) {        // 5 async issues/thread
    const int s   = i * NTHREADS + tid;
    const int r   = s >> 7;   // staged row 0..9
    const int seg = s & 127;  // float4 segment 0..127
    int gy = y0 - 1 + r;
    gy = gy < 0 ? 0 : (gy > H - 1 ? H - 1 : gy);              // replicate top/bottom
    const uint64_t gaddr = (uint64_t)(uintptr_t)(plane + (size_t)gy * W + seg * 4);
    const uint32_t laddr = tile_base + (uint32_t)((r * LDSW + 4 + seg * 4) * 4);
    asm volatile("global_load_async_to_lds_b128 %0, %1, off"
                 :: "v"(laddr), "v"(gaddr) : "memory");
  }
  asm volatile("s_wait_asynccnt 0x0" ::: "memory");           // my wave's DMAs done
  __syncthreads();                                            // everyone's DMAs visible

  // Replicate left/right halo columns.
  if (tid < ROWS) {
    tile[tid * LDSW + 3]     = tile[tid * LDSW + 4];          // x = -1
    tile[tid * LDSW + 4 + W] = tile[tid * LDSW + 3 + W];      // x = 512
  }
  __syncthreads();

  // ---- Compute 8 x 512 medians; each thread emits 4 float4s (contiguous per wave).
  #pragma unroll
  for (int i = 0; i < (TILE_H * SEGS) / NTHREADS; ++i) {      // 4 iterations
    const int idx = i * NTHREADS + tid;                       // 0..1023
    const int oy  = idx >> 7;                                 // tile-local out row
    const int x4  = idx & 127;                                // float4 column
    const int c0  = 3 + x4 * 4;                               // LDS col of x-1

    const float* r0 = &tile[(oy + 0) * LDSW + c0];
    const float* r1 = &tile[(oy + 1) * LDSW + c0];
    const float* r2 = &tile[(oy + 2) * LDSW + c0];

    float lo[6], md[6], hi[6];
    #pragma unroll
    for (int j = 0; j < 6; ++j)
      sort3(r0[j], r1[j], r2[j], lo[j], md[j], hi[j]);

    v4f o;
    #pragma unroll
    for (int j = 0; j < 4; ++j) {
      const float A = fmaxf(fmaxf(lo[j], lo[j + 1]), lo[j + 2]);  // max of column mins
      const float B = med3(md[j], md[j + 1], md[j + 2]);          // med of column meds
      const float C = fminf(fminf(hi[j], hi[j + 1]), hi[j + 2]);  // min of column maxes
      o[j] = med3(A, B, C);                                       // exact 3x3 median
    }

    const int gy = y0 + oy;
    v4f* dst = (v4f*)(out + ((size_t)bc * H + gy) * W + x4 * 4);
    __builtin_nontemporal_store(o, dst);   // write-once output: keep L2 for input reuse
  }
}

extern "C" void kernel_launch(void* const* d_in, const int* in_sizes, int n_in,
                              void* d_out, int out_size, void* d_ws, size_t ws_size,
                              hipStream_t stream) {
  const float* noised = (const float*)d_in[0];
  const float* cover  = (const float*)d_in[1];
  float* out = (float*)d_out;

  const int N      = in_sizes[0];        // 32*3*512*512 = 25,165,824
  const int planes = N / (H * W);        // 96

  MedianBlur3x3_kernel<<<dim3(planes * TILESY), dim3(NTHREADS), 0, stream>>>(noised, out);

  // cover_image pass-through: pure D2D copy at peak bandwidth (graph-capturable).
  hipMemcpyAsync(out + N, cover, (size_t)N * sizeof(float),
                 hipMemcpyDeviceToDevice, stream);
}